// NetVLAD_45346264711389
// MI455X (gfx1250) — compile-verified
//
#include <hip/hip_runtime.h>
#include <math.h>

#define NN 32
#define CC 512
#define PP 1600
#define KK 64
#define PB 160
#define PBLOCKS 10   // PP / PB
#define STEPS 10     // PB / 16

#define XT_STRIDE 516   // 16 x-rows [p][c], stride 516 -> bank-conflict-free, even (b64 ok)
#define SP_STRIDE 18    // score/soft rows [k][p], stride 18 -> even (b64 ok)

typedef float v2f __attribute__((ext_vector_type(2)));
typedef float v8f __attribute__((ext_vector_type(8)));

__device__ __forceinline__ void atomic_add_f32(float* p, float v) {
    __hip_atomic_fetch_add(p, v, __ATOMIC_RELAXED, __HIP_MEMORY_SCOPE_AGENT);
}

// -------------------------------------------------------------------------
// Kernel 1: fused  scores = conv_w @ x + b  -> softmax(k) -> agg += soft @ x^T
// grid: NN*PBLOCKS blocks of 512 threads (16 waves).
// wave w: kt = w%4 (16 clusters), cq = w/4 (128 channels).
// -------------------------------------------------------------------------
__global__ __launch_bounds__(512)
void nv_gemm_softmax_agg(const float* __restrict__ x,
                         const float* __restrict__ conv_w,
                         const float* __restrict__ conv_b,
                         float* __restrict__ agg,
                         float* __restrict__ mass)
{
    __shared__ float xT[16 * XT_STRIDE];        // x tile, transposed: xT[p][c]
    __shared__ float part[4 * KK * SP_STRIDE];  // per-cq score partials; q=0 reused as soft

    const int n    = blockIdx.x / PBLOCKS;
    const int pb   = blockIdx.x % PBLOCKS;
    const int tid  = threadIdx.x;
    const int wave = tid >> 5;
    const int lane = tid & 31;
    const int lhalf = lane >> 4;    // 0 or 1
    const int lmod  = lane & 15;
    const int kbase = (wave & 3) * 16;
    const int cq    = wave >> 2;
    const int cbase = cq * 128;

    // conv_w A-fragments, resident in registers for the whole kernel.
    // A layout: lane holds A[M = lane%16][K = (lane/16)*2 + v]
    v2f aw[32];
#pragma unroll
    for (int cs = 0; cs < 32; ++cs) {
        const int col = cbase + cs * 4 + lhalf * 2;
        aw[cs] = *(const v2f*)(conv_w + (size_t)(kbase + lmod) * CC + col);
    }

    v8f acc[8] = {};        // agg accumulators: 8 c-tiles of 16x16
    float m_acc = 0.0f;     // mass accumulator (threads 0..63 only)

    const float* xn = x + (size_t)n * CC * PP;

    for (int s = 0; s < STEPS; ++s) {
        const int pbase = pb * PB + s * 16;

        // ---- stage x[:, pbase:pbase+16] into LDS as xT[p][c] ----
#pragma unroll
        for (int rep = 0; rep < 16; ++rep) {
            const int i = rep * 512 + tid;     // 8192 floats
            const int c = i >> 4;
            const int p = i & 15;
            xT[p * XT_STRIDE + c] = xn[(size_t)c * PP + pbase + p];
        }
        __syncthreads();

        // ---- GEMM1 partial: D[16k x 16p] over this wave's 128 channels ----
        v8f d = {};
#pragma unroll
        for (int cs = 0; cs < 32; ++cs) {
            // B layout: lane holds B[K=(lane/16)*2+v][N=lane%16]; K-dim = channel
            const v2f bx = *(const v2f*)(&xT[lmod * XT_STRIDE + cbase + cs * 4 + lhalf * 2]);
            d = __builtin_amdgcn_wmma_f32_16x16x4_f32(false, aw[cs], false, bx,
                                                      (short)0, d, false, false);
        }
        // C/D layout: lane l, vgpr r -> k = kbase + r + 8*(l/16), p = l%16
#pragma unroll
        for (int r = 0; r < 8; ++r) {
            const int k = kbase + r + 8 * lhalf;
            part[(cq * KK + k) * SP_STRIDE + lmod] = d[r];
        }
        __syncthreads();

        // ---- softmax over k, one thread per pixel ----
        if (tid < 16) {
            const int p = tid;
            float mx = -3.4e38f;
            for (int k = 0; k < KK; ++k) {
                float sv = part[k * SP_STRIDE + p]
                         + part[(KK     + k) * SP_STRIDE + p]
                         + part[(2 * KK + k) * SP_STRIDE + p]
                         + part[(3 * KK + k) * SP_STRIDE + p]
                         + conv_b[k];
                part[k * SP_STRIDE + p] = sv;
                mx = fmaxf(mx, sv);
            }
            float sum = 0.0f;
            for (int k = 0; k < KK; ++k) {
                const float e = __expf(part[k * SP_STRIDE + p] - mx);
                sum += e;
                part[k * SP_STRIDE + p] = e;
            }
            const float inv = 1.0f / sum;
            for (int k = 0; k < KK; ++k) part[k * SP_STRIDE + p] *= inv;
        }
        __syncthreads();

        // ---- mass: one thread per cluster sums soft over the 16 pixels ----
        if (tid < KK) {
            float a = 0.0f;
#pragma unroll
            for (int p = 0; p < 16; ++p) a += part[tid * SP_STRIDE + p];
            m_acc += a;
        }

        // ---- GEMM2: acc[k,c] += soft[k,p] * x[c,p]  (K-dim = pixel) ----
        v2f as[4];
#pragma unroll
        for (int ps = 0; ps < 4; ++ps)
            as[ps] = *(const v2f*)(&part[(kbase + lmod) * SP_STRIDE + ps * 4 + lhalf * 2]);
#pragma unroll
        for (int ct = 0; ct < 8; ++ct) {
#pragma unroll
            for (int ps = 0; ps < 4; ++ps) {
                const int prow = ps * 4 + lhalf * 2;
                v2f bx;
                bx.x = xT[prow * XT_STRIDE + cbase + ct * 16 + lmod];
                bx.y = xT[(prow + 1) * XT_STRIDE + cbase + ct * 16 + lmod];
                acc[ct] = __builtin_amdgcn_wmma_f32_16x16x4_f32(false, as[ps], false, bx,
                                                                (short)0, acc[ct], false, false);
            }
        }
        __syncthreads();   // protect xT/part before next step's writes
    }

    // ---- epilogue: atomically accumulate block partials ----
    if (tid < KK) atomic_add_f32(&mass[n * KK + tid], m_acc);

    float* aggn = agg + (size_t)n * KK * CC;
#pragma unroll
    for (int ct = 0; ct < 8; ++ct) {
#pragma unroll
        for (int r = 0; r < 8; ++r) {
            const int k = kbase + r + 8 * lhalf;
            const int c = cbase + ct * 16 + lmod;
            atomic_add_f32(&aggn[(size_t)k * CC + c], acc[ct][r]);
        }
    }
}

// -------------------------------------------------------------------------
// Kernel 2: vlad = agg - mass*centroid, intra-normalize over C (one block per n,k)
// -------------------------------------------------------------------------
__global__ __launch_bounds__(256)
void nv_vlad_intranorm(const float* __restrict__ agg,
                       const float* __restrict__ mass,
                       const float* __restrict__ cent,
                       float* __restrict__ out,
                       float* __restrict__ gnorm)
{
    const int b = blockIdx.x;     // n*64 + k
    const int n = b >> 6;
    const int k = b & 63;
    const int tid = threadIdx.x;
    const size_t base = (size_t)b * CC;

    const float mk = mass[b];
    float v0 = agg[base + tid]       - mk * cent[k * CC + tid];
    float v1 = agg[base + 256 + tid] - mk * cent[k * CC + 256 + tid];
    float ss = v0 * v0 + v1 * v1;

#pragma unroll
    for (int off = 16; off > 0; off >>= 1) ss += __shfl_xor(ss, off, 32);

    __shared__ float red[8];
    __shared__ float stot;
    if ((tid & 31) == 0) red[tid >> 5] = ss;
    __syncthreads();
    if (tid == 0) {
        float sacc = 0.0f;
#pragma unroll
        for (int i = 0; i < 8; ++i) sacc += red[i];
        stot = sacc;
    }
    __syncthreads();

    const float s   = stot;
    const float inv = 1.0f / fmaxf(sqrtf(s), 1e-12f);
    out[base + tid]       = v0 * inv;
    out[base + 256 + tid] = v1 * inv;
    if (tid == 0) atomic_add_f32(&gnorm[n], s * inv * inv);
}

// -------------------------------------------------------------------------
// Kernel 3: global L2 normalize per image
// -------------------------------------------------------------------------
__global__ __launch_bounds__(256)
void nv_globalnorm(float* __restrict__ out, const float* __restrict__ gnorm)
{
    const int i = blockIdx.x * 256 + threadIdx.x;
    const int n = i >> 15;   // KK*CC = 32768 per image
    out[i] *= 1.0f / fmaxf(sqrtf(gnorm[n]), 1e-12f);
}

// -------------------------------------------------------------------------
extern "C" void kernel_launch(void* const* d_in, const int* in_sizes, int n_in,
                              void* d_out, int out_size, void* d_ws, size_t ws_size,
                              hipStream_t stream)
{
    const float* x      = (const float*)d_in[0];   // [N, C, H, W]
    const float* cent   = (const float*)d_in[1];   // [K, C]
    const float* conv_w = (const float*)d_in[2];   // [K, C]
    const float* conv_b = (const float*)d_in[3];   // [K]
    float* out = (float*)d_out;                    // [N, K*C]

    float* agg  = (float*)d_ws;                    // N*K*C floats = 4 MB
    float* mass = agg + (size_t)NN * KK * CC;      // N*K floats
    float* gn   = mass + NN * KK;                  // N floats

    const size_t zero_bytes = ((size_t)NN * KK * CC + NN * KK + NN) * sizeof(float);
    hipMemsetAsync(d_ws, 0, zero_bytes, stream);

    nv_gemm_softmax_agg<<<NN * PBLOCKS, 512, 0, stream>>>(x, conv_w, conv_b, agg, mass);
    nv_vlad_intranorm<<<NN * KK, 256, 0, stream>>>(agg, mass, cent, out, gn);
    nv_globalnorm<<<(NN * KK * CC) / 256, 256, 0, stream>>>(out, gn);
}